// MultViewPoolingOctreeEncoder_76948634075540
// MI455X (gfx1250) — compile-verified
//
#include <hip/hip_runtime.h>
#include <math.h>

// ---------------- problem constants ----------------
#define VIEWS     6
#define PV        89600                 // points per view = 64*28*50
#define P_TOTAL   (VIEWS * PV)          // 537600 points
#define C_ALL     65                    // 64 features + density
#define C_OUT     33                    // 32 features + density per level
#define RES8      256
#define RES7      128
#define W8        (1 << 19)             // 256^3 / 32 bitmap words (2 MB)
#define W7        (1 << 16)             // 128^3 / 32 bitmap words (256 KB)

// ---------------- tuning ----------------
#define TILE      128                   // points per scatter block (divides PV: 89600 = 700*128)
#define SCAN_T    256
#define WPW       8                     // words per thread in scan (2048 words/block)

// ---------------- CDNA5 async global->LDS helpers ----------------
typedef int v4i __attribute__((vector_size(16)));
typedef __attribute__((address_space(1))) v4i* g_v4i_p;   // global
typedef __attribute__((address_space(3))) v4i* l_v4i_p;   // LDS

__device__ __forceinline__ void async_load_b128(const float* g, float* l) {
#if defined(__has_builtin) && __has_builtin(__builtin_amdgcn_global_load_async_to_lds_b128)
  __builtin_amdgcn_global_load_async_to_lds_b128((g_v4i_p)g, (l_v4i_p)l, 0, 0);
#else
  unsigned ldsoff = (unsigned)(unsigned long long)(void*)l;
  asm volatile("global_load_async_to_lds_b128 %0, %1, off"
               :: "v"(ldsoff), "v"(g) : "memory");
#endif
}

__device__ __forceinline__ void wait_async0() {
#if defined(__has_builtin) && __has_builtin(__builtin_amdgcn_s_wait_asynccnt)
  __builtin_amdgcn_s_wait_asynccnt(0);
#else
  asm volatile("s_wait_asynccnt 0" ::: "memory");
#endif
}

// ---------------- math helpers ----------------
__device__ __forceinline__ float fast_tanh(float x) {
#if defined(__has_builtin) && __has_builtin(__builtin_amdgcn_tanhf)
  return __builtin_amdgcn_tanhf(x);       // gfx1250 native V_TANH_F32
#else
  return tanhf(x);
#endif
}

__device__ __forceinline__ unsigned vox_id(float s0, float s1, float s2, int res) {
  int q0 = (int)((s0 * 0.5f + 0.5f) * (float)res);
  int q1 = (int)((s1 * 0.5f + 0.5f) * (float)res);
  int q2 = (int)((s2 * 0.5f + 0.5f) * (float)res);
  q0 = min(max(q0, 0), res - 1);
  q1 = min(max(q1, 0), res - 1);
  q2 = min(max(q2, 0), res - 1);
  return ((unsigned)q0 * (unsigned)res + (unsigned)q1) * (unsigned)res + (unsigned)q2;
}

__device__ __forceinline__ unsigned rank_of(const unsigned* __restrict__ bm,
                                            const unsigned* __restrict__ pfx,
                                            unsigned id) {
  unsigned w = id >> 5, b = id & 31u;
  return pfx[w] + (unsigned)__popc(bm[w] & ((1u << b) - 1u));
}

// ---------------- phase 1: presence bitmaps + id precompute ----------------
__global__ void mark_kernel(const float* __restrict__ coords,
                            unsigned* __restrict__ bm8, unsigned* __restrict__ bm7,
                            unsigned* __restrict__ ids8, unsigned* __restrict__ ids7) {
  int p = blockIdx.x * blockDim.x + threadIdx.x;
  if (p >= P_TOTAL) return;
  float x = coords[3 * p + 0], y = coords[3 * p + 1], z = coords[3 * p + 2];
  float s0 = fast_tanh(x * (1.0f / 50.0f));
  float s1 = fast_tanh(y * (1.0f / 50.0f));
  float s2 = fast_tanh(z * (1.0f / 10.0f));
  unsigned id8 = vox_id(s0, s1, s2, RES8);
  unsigned id7 = vox_id(s0, s1, s2, RES7);
  ids8[p] = id8;
  ids7[p] = id7;
  atomicOr(&bm8[id8 >> 5], 1u << (id8 & 31u));
  atomicOr(&bm7[id7 >> 5], 1u << (id7 & 31u));
}

// ---------------- phase 2: popcount prefix scan over bitmap words ----------------
__global__ void block_sums_kernel(const unsigned* __restrict__ bm,
                                  unsigned* __restrict__ bsums) {
  __shared__ unsigned sh[SCAN_T];
  int t = threadIdx.x;
  int base = blockIdx.x * SCAN_T * WPW + t * WPW;
  unsigned s = 0;
#pragma unroll
  for (int k = 0; k < WPW; ++k) s += (unsigned)__popc(bm[base + k]);
  sh[t] = s;
  __syncthreads();
  for (int off = SCAN_T / 2; off > 0; off >>= 1) {
    if (t < off) sh[t] += sh[t + off];
    __syncthreads();
  }
  if (t == 0) bsums[blockIdx.x] = sh[0];
}

__global__ void scan_small_kernel(unsigned* __restrict__ bsums, int n) {
  __shared__ unsigned sh[SCAN_T];
  int t = threadIdx.x;
  unsigned v = (t < n) ? bsums[t] : 0u;
  sh[t] = v;
  __syncthreads();
  for (int off = 1; off < SCAN_T; off <<= 1) {
    unsigned u = (t >= off) ? sh[t - off] : 0u;
    __syncthreads();
    sh[t] += u;
    __syncthreads();
  }
  if (t < n) bsums[t] = sh[t] - v;   // exclusive
}

__global__ void write_prefix_kernel(const unsigned* __restrict__ bm,
                                    const unsigned* __restrict__ bsums,
                                    unsigned* __restrict__ pfx) {
  __shared__ unsigned sh[SCAN_T];
  int t = threadIdx.x;
  int base = blockIdx.x * SCAN_T * WPW + t * WPW;
  unsigned local[WPW];
  unsigned s = 0;
#pragma unroll
  for (int k = 0; k < WPW; ++k) {
    local[k] = (unsigned)__popc(bm[base + k]);
    s += local[k];
  }
  sh[t] = s;
  __syncthreads();
  unsigned v = s;
  for (int off = 1; off < SCAN_T; off <<= 1) {
    unsigned u = (t >= off) ? sh[t - off] : 0u;
    __syncthreads();
    sh[t] += u;
    __syncthreads();
  }
  unsigned run = bsums[blockIdx.x] + sh[t] - v;  // exclusive word prefix
#pragma unroll
  for (int k = 0; k < WPW; ++k) {
    pfx[base + k] = run;
    run += local[k];
  }
}

// ---------------- phase 3: ids -> ranks (in place) + per-voxel counts ----------------
__global__ void count_rank_kernel(unsigned* __restrict__ ids8, unsigned* __restrict__ ids7,
                                  const unsigned* __restrict__ bm8, const unsigned* __restrict__ pfx8,
                                  const unsigned* __restrict__ bm7, const unsigned* __restrict__ pfx7,
                                  float* __restrict__ cnt) {
  int p = blockIdx.x * blockDim.x + threadIdx.x;
  if (p >= P_TOTAL) return;
  unsigned r8 = rank_of(bm8, pfx8, ids8[p]);
  unsigned r7 = rank_of(bm7, pfx7, ids7[p]);
  ids8[p] = r8;                          // overwrite id with rank
  ids7[p] = r7;
  atomicAdd(&cnt[r8], 1.0f);
  atomicAdd(&cnt[P_TOTAL + r7], 1.0f);
}

// ---------------- phase 4: async-staged, pre-scaled feature scatter ----------------
__global__ void __launch_bounds__(TILE)
scatter_kernel(const float* __restrict__ es,
               const unsigned* __restrict__ rank8, const unsigned* __restrict__ rank7,
               const float* __restrict__ cnt, float* __restrict__ out) {
  __shared__ float tile[C_ALL * TILE];                    // 65*128*4 = 33280 B
  const int tid   = threadIdx.x;
  const int view  = blockIdx.x / (PV / TILE);             // 0..5
  const int tbase = (blockIdx.x % (PV / TILE)) * TILE;    // point base within view
  const float* es_view = es + (long long)view * C_ALL * PV;

  // Stage 65 contiguous feature rows of TILE floats each via async b128 copies.
  const int NXFER = C_ALL * (TILE / 4);                   // 2080
  for (int i = tid; i < NXFER; i += TILE) {
    int c    = i / (TILE / 4);
    int col4 = (i % (TILE / 4)) * 4;
    async_load_b128(es_view + (long long)c * PV + tbase + col4,
                    &tile[c * TILE + col4]);
  }

  // Overlap rank/count loads with the async staging.
  int p = view * PV + tbase + tid;
  unsigned r8 = rank8[p];
  unsigned r7 = rank7[p];
  float inv8 = 1.0f / cnt[r8];            // cnt >= 1 for occupied ranks
  float inv7 = 1.0f / cnt[P_TOTAL + r7];

  wait_async0();
  __syncthreads();

  float dens = tile[64 * TILE + tid];
  float* out0 = out + (size_t)r8 * C_OUT;                            // level 8 plane
  float* out1 = out + (size_t)P_TOTAL * C_OUT + (size_t)r7 * C_OUT;  // level 7 plane
#pragma unroll 4
  for (int c = 0; c < 32; ++c) {
    atomicAdd(&out0[c], tile[c * TILE + tid] * inv8);
    atomicAdd(&out1[c], tile[(32 + c) * TILE + tid] * inv7);
  }
  atomicAdd(&out0[32], dens * inv8);
  atomicAdd(&out1[32], dens * inv7);
}

// ---------------- launcher ----------------
extern "C" void kernel_launch(void* const* d_in, const int* in_sizes, int n_in,
                              void* d_out, int out_size, void* d_ws, size_t ws_size,
                              hipStream_t stream) {
  (void)in_sizes; (void)n_in; (void)ws_size;
  const float* es     = (const float*)d_in[0];   // [6,65,64,28,50]
  const float* coords = (const float*)d_in[1];   // [1,6,64,28,50,3]
  float* out = (float*)d_out;                    // [2, P, 33]

  // workspace layout (u32 units); zeroed region first
  unsigned* bm8  = (unsigned*)d_ws;              // W8
  unsigned* bm7  = bm8  + W8;                    // W7
  float*    cnt  = (float*)(bm7 + W7);           // 2*P
  unsigned* pfx8 = (unsigned*)(cnt + 2 * P_TOTAL); // W8
  unsigned* pfx7 = pfx8 + W8;                    // W7
  unsigned* bs8  = pfx7 + W7;                    // 256
  unsigned* bs7  = bs8  + 256;                   // 256
  unsigned* ids8 = bs7  + 256;                   // P
  unsigned* ids7 = ids8 + P_TOTAL;               // P
  size_t zero_bytes = ((size_t)W8 + W7 + 2 * P_TOTAL) * sizeof(unsigned);

  (void)hipMemsetAsync(d_ws, 0, zero_bytes, stream);
  (void)hipMemsetAsync(d_out, 0, (size_t)out_size * sizeof(float), stream);

  mark_kernel<<<P_TOTAL / 256, 256, 0, stream>>>(coords, bm8, bm7, ids8, ids7);

  block_sums_kernel<<<W8 / (SCAN_T * WPW), SCAN_T, 0, stream>>>(bm8, bs8);
  scan_small_kernel<<<1, SCAN_T, 0, stream>>>(bs8, W8 / (SCAN_T * WPW));
  write_prefix_kernel<<<W8 / (SCAN_T * WPW), SCAN_T, 0, stream>>>(bm8, bs8, pfx8);

  block_sums_kernel<<<W7 / (SCAN_T * WPW), SCAN_T, 0, stream>>>(bm7, bs7);
  scan_small_kernel<<<1, SCAN_T, 0, stream>>>(bs7, W7 / (SCAN_T * WPW));
  write_prefix_kernel<<<W7 / (SCAN_T * WPW), SCAN_T, 0, stream>>>(bm7, bs7, pfx7);

  count_rank_kernel<<<P_TOTAL / 256, 256, 0, stream>>>(ids8, ids7,
                                                       bm8, pfx8, bm7, pfx7, cnt);

  scatter_kernel<<<VIEWS * (PV / TILE), TILE, 0, stream>>>(es, ids8, ids7, cnt, out);
}